// CRFLoss_61795989454922
// MI455X (gfx1250) — compile-verified
//
#include <hip/hip_runtime.h>
#include <stdint.h>

typedef __attribute__((ext_vector_type(2))) float v2f;
typedef __attribute__((ext_vector_type(8))) float v8f;

#define Bn 64
#define Tn 8192
#define Cn 48
#define TILE_T 128
#define PITCH 52   // dwords per LDS row: conflict-free & keeps b64/b128 alignment

// ---------------- init: zero the double accumulator (ws is poisoned) --------
__global__ void crf_init(double* accum) {
    if (threadIdx.x == 0) accum[0] = 0.0;
}

// ---------------- tiny: log_softmax of A_start (48) and A_trans (48x49) -----
__global__ void crf_lsm(const float* __restrict__ A_start,
                        const float* __restrict__ A_trans,
                        float* __restrict__ lsm0,
                        float* __restrict__ lsmA) {
    int r = threadIdx.x;
    if (r < Cn) {
        const float* row = A_trans + r * (Cn + 1);
        float m = row[0];
        for (int c = 1; c <= Cn; ++c) m = fmaxf(m, row[c]);
        float s = 0.f;
        for (int c = 0; c <= Cn; ++c) s += expf(row[c] - m);
        float lz = m + logf(s);
        float* orow = lsmA + r * (Cn + 1);
        for (int c = 0; c <= Cn; ++c) orow[c] = row[c] - lz;
    } else if (r == Cn) {
        float m = A_start[0];
        for (int c = 1; c < Cn; ++c) m = fmaxf(m, A_start[c]);
        float s = 0.f;
        for (int c = 0; c < Cn; ++c) s += expf(A_start[c] - m);
        float lz = m + logf(s);
        for (int c = 0; c < Cn; ++c) lsm0[c] = A_start[c] - lz;
    }
}

// ---------------- main: streaming reduction over (B,T) ----------------------
__global__ __launch_bounds__(256) void crf_main(
    const float* __restrict__ lp, const int* __restrict__ lens,
    const int* __restrict__ labels, const float* __restrict__ lsm0,
    const float* __restrict__ lsmA, double* __restrict__ accum) {

    __shared__ float tile[TILE_T * PITCH];
    __shared__ float wsum[8];

    const int tid  = threadIdx.x;
    const int lane = tid & 31;
    const int wv   = tid >> 5;          // wave 0..7, owns 16 rows
    const int b    = blockIdx.y;
    const int t0   = blockIdx.x * TILE_T;
    const int len  = lens[b];
    const int r0   = wv * 16;           // first row (timestep) of this wave

    // ---- CDNA5 async global->LDS copy of this wave's 16 rows (16x48 f32) ---
    const float* gbase = lp + ((size_t)b * Tn + (size_t)(t0 + r0)) * Cn;
    uint32_t lds_base = (uint32_t)(uintptr_t)(&tile[r0 * PITCH]);
    #pragma unroll
    for (int i = 0; i < 6; ++i) {           // 192 b128 quads / 32 lanes
        int q   = lane + 32 * i;            // 0..191
        int row = q / 12;                   // 0..15
        int cq  = q - row * 12;             // 0..11 (quad of 4 floats)
        unsigned long long ga = (unsigned long long)(uintptr_t)(gbase + row * Cn + cq * 4);
        uint32_t lo = lds_base + (uint32_t)((row * PITCH + cq * 4) * 4);
        asm volatile("global_load_async_to_lds_b128 %0, %1, off"
                     :: "v"(lo), "v"(ga) : "memory");
    }
    asm volatile("s_wait_asynccnt 0" ::: "memory");

    // ---- WMMA ones-matmul: per-row sum of exp(lp[c]) for c=1..47 -----------
    // A (16x4 f32): lane<16 holds (M=lane, K=0..1), lane>=16 holds (M=lane-16, K=2..3)
    const int M  = lane & 15;
    const int kh = lane >> 4;               // 0 or 1
    const float* lrow = &tile[(r0 + M) * PITCH + 2 * kh];
    v2f bones; bones.x = 1.0f; bones.y = 1.0f;    // all-ones B: D[m][n] = row-sum
    v8f acc = {};
    float Y = 0.0f;                          // lp[t][0] (class I_ID=0), lanes<16
    #pragma unroll
    for (int j = 0; j < 12; ++j) {           // 12 chunks of K=4 cover C=48
        float p0 = lrow[4 * j + 0];
        float p1 = lrow[4 * j + 1];
        bool isC0 = (j == 0) && (kh == 0);   // class 0 excluded from the sum
        if (isC0) Y = p0;
        v2f a;
        a.x = isC0 ? 0.0f : __expf(p0);
        a.y = __expf(p1);
        acc = __builtin_amdgcn_wmma_f32_16x16x4_f32(
                  false, a, false, bones, (short)0, acc, false, false);
    }

    // ---- pull row M's sum to lane M: D row M sits in VGPR (M&7) of lane 0/16
    float rs = 0.0f;
    #pragma unroll
    for (int r = 0; r < 8; ++r) {
        float v = __shfl(acc[r], (lane & 8) ? 16 : 0, 32);
        if ((lane & 7) == r) rs = v;
    }

    // ---- per-timestep contribution (lanes 0..15 own rows r0..r0+15) --------
    float contrib = 0.0f;
    if (lane < 16) {
        int t = t0 + r0 + lane;
        if (t < len) {
            int y = labels[(size_t)b * Tn + t];            // valid: t < len
            float S    = logf(rs);                         // logsumexp(lp[1:])
            float emis = tile[(r0 + lane) * PITCH + y];
            float tr   = (t == 0) ? lsm0[y]
                                  : lsmA[labels[(size_t)b * Tn + t - 1] * (Cn + 1) + y];
            float den, fin = 0.0f;
            if (t == len - 1) {
                den = S;
                fin = lsmA[y * (Cn + 1) + Cn];             // lsmA[y_last, C]
            } else {
                float mx = fmaxf(S, Y), mn = fminf(S, Y);  // logaddexp(S, Y)
                den = mx + log1pf(expf(mn - mx));
            }
            contrib = emis + tr + fin - den;
        }
    }

    // ---- block reduce -> one f64 atomic per block --------------------------
    #pragma unroll
    for (int o = 16; o > 0; o >>= 1) contrib += __shfl_down(contrib, o, 32);
    if (lane == 0) wsum[wv] = contrib;
    __syncthreads();
    if (tid == 0) {
        float bs = 0.0f;
        for (int i = 0; i < 8; ++i) bs += wsum[i];
        atomicAdd(accum, (double)bs);                      // global_atomic_add_f64
    }
}

// ---------------- finalize: divide by total token count ---------------------
__global__ void crf_fin(const double* accum, const int* __restrict__ lens, float* out) {
    if (threadIdx.x == 0) {
        long long nt = 0;
        for (int i = 0; i < Bn; ++i) nt += lens[i];
        out[0] = (float)(accum[0] / (double)nt);
    }
}

extern "C" void kernel_launch(void* const* d_in, const int* in_sizes, int n_in,
                              void* d_out, int out_size, void* d_ws, size_t ws_size,
                              hipStream_t stream) {
    const float* log_probs  = (const float*)d_in[0];
    const int*   input_lens = (const int*)  d_in[1];
    const int*   labels     = (const int*)  d_in[2];
    const float* A_start    = (const float*)d_in[3];
    const float* A_trans    = (const float*)d_in[4];
    float* out = (float*)d_out;

    double* accum = (double*)d_ws;
    float*  lsm0  = (float*)((char*)d_ws + 16);
    float*  lsmA  = (float*)((char*)d_ws + 16 + Cn * 4);

    crf_init<<<1, 64, 0, stream>>>(accum);
    crf_lsm <<<1, 64, 0, stream>>>(A_start, A_trans, lsm0, lsmA);
    crf_main<<<dim3(Tn / TILE_T, Bn), 256, 0, stream>>>(log_probs, input_lens,
                                                        labels, lsm0, lsmA, accum);
    crf_fin <<<1, 64, 0, stream>>>(accum, input_lens, out);
}